// mem_update_9251359556235
// MI455X (gfx1250) — compile-verified
//
#include <hip/hip_runtime.h>

// LIF membrane/spike scan over T=4 timesteps, fp32 in -> fp32 (0/1) out.
// Memory-bound streaming kernel: 411 MB of HBM traffic, ~17.6 us roofline
// at 23.3 TB/s. 128-bit nontemporal loads/stores (working set has zero reuse
// and is > 2x the 192 MB L2, so NT cache policy is the right TH hint).

typedef float v4f __attribute__((ext_vector_type(4)));

static constexpr float kThresh = 0.5f;
static constexpr float kDecay  = 0.25f;

// One scan step, branchless:
//   spike_out = (m > thresh) ? 1 : 0
//   m_next    = m * (spike ? 0 : decay) + x_next   (one v_cmp, two v_cndmask, one v_fma)
__device__ __forceinline__ void lif_step(float& m, float x_next, float& spike_out) {
  const bool fired = m > kThresh;
  spike_out = fired ? 1.0f : 0.0f;
  const float g = fired ? 0.0f : kDecay;
  m = __builtin_fmaf(m, g, x_next);
}

__global__ __launch_bounds__(256) void lif_scan4_vec4(const float* __restrict__ x,
                                                      float* __restrict__ out,
                                                      unsigned n4) {
  const unsigned i = blockIdx.x * blockDim.x + threadIdx.x;
  if (i >= n4) return;

  const v4f* __restrict__ xv = (const v4f*)x;
  v4f* __restrict__ ov = (v4f*)out;
  const size_t st = (size_t)n4;  // timestep stride in float4 units

  // Issue all 4 independent 128-bit NT loads up front (4 HBM streams in flight).
  v4f x0 = __builtin_nontemporal_load(xv + 0 * st + i);
  v4f x1 = __builtin_nontemporal_load(xv + 1 * st + i);
  v4f x2 = __builtin_nontemporal_load(xv + 2 * st + i);
  v4f x3 = __builtin_nontemporal_load(xv + 3 * st + i);

  v4f s0 = {}, s1 = {}, s2 = {}, s3 = {};
#pragma unroll
  for (int c = 0; c < 4; ++c) {
    float m = x0[c];           // step 0: mem = x[0]
    float sp;
    lif_step(m, x1[c], sp); s0[c] = sp;
    lif_step(m, x2[c], sp); s1[c] = sp;
    lif_step(m, x3[c], sp); s2[c] = sp;
    s3[c] = (m > kThresh) ? 1.0f : 0.0f;  // last spike, no mem update needed
  }

  __builtin_nontemporal_store(s0, ov + 0 * st + i);
  __builtin_nontemporal_store(s1, ov + 1 * st + i);
  __builtin_nontemporal_store(s2, ov + 2 * st + i);
  __builtin_nontemporal_store(s3, ov + 3 * st + i);
}

// Scalar fallback for plane sizes not divisible by 4 (not hit for the harness
// shape 32*128*56*56, but keeps kernel_launch correct for any size).
__global__ __launch_bounds__(256) void lif_scan4_scalar(const float* __restrict__ x,
                                                        float* __restrict__ out,
                                                        unsigned start, unsigned n) {
  const unsigned i = start + blockIdx.x * blockDim.x + threadIdx.x;
  if (i >= n) return;
  const size_t st = (size_t)n;

  float m = __builtin_nontemporal_load(x + 0 * st + i);
  float sp;
  float o0, o1, o2, o3;
  lif_step(m, __builtin_nontemporal_load(x + 1 * st + i), sp); o0 = sp;
  lif_step(m, __builtin_nontemporal_load(x + 2 * st + i), sp); o1 = sp;
  lif_step(m, __builtin_nontemporal_load(x + 3 * st + i), sp); o2 = sp;
  o3 = (m > kThresh) ? 1.0f : 0.0f;

  __builtin_nontemporal_store(o0, out + 0 * st + i);
  __builtin_nontemporal_store(o1, out + 1 * st + i);
  __builtin_nontemporal_store(o2, out + 2 * st + i);
  __builtin_nontemporal_store(o3, out + 3 * st + i);
}

extern "C" void kernel_launch(void* const* d_in, const int* in_sizes, int n_in,
                              void* d_out, int out_size, void* d_ws, size_t ws_size,
                              hipStream_t stream) {
  const float* x = (const float*)d_in[0];
  float* out = (float*)d_out;

  const unsigned T = 4;                                // scan is unrolled for T=4
  const unsigned total = (unsigned)in_sizes[0];        // 51,380,224
  const unsigned n = total / T;                        // elements per timestep plane

  if ((n & 3u) == 0u) {
    const unsigned n4 = n >> 2;                        // float4 columns
    const unsigned blocks = (n4 + 255u) / 256u;
    lif_scan4_vec4<<<blocks, 256, 0, stream>>>(x, out, n4);
  } else {
    const unsigned blocks = (n + 255u) / 256u;
    lif_scan4_scalar<<<blocks, 256, 0, stream>>>(x, out, 0u, n);
  }
}